// LocalityEntropyLoss_74002286510759
// MI455X (gfx1250) — compile-verified
//
#include <hip/hip_runtime.h>

// LocalityEntropyLoss for MI455X (gfx1250, wave32).
// feat_map: [256, 14*14*2048] f32.  Outputs: d_out[0]=L1 (mean row entropy),
// d_out[1]=L2 (= -entropy(column mean)).
//
// Memory-bound: two full streaming passes (~822 MB @ 23.3 TB/s ~= 35 us).
// Online softmax makes each pass single-read. WMMA f32 16x16x4 is used for
// the final mean-of-256-entropies (matmul-by-ones), wave32 reductions +
// global_prefetch_b8 in the hot loops.

#define T_ROWS   256
#define FEAT     401408          // 14*14*2048
#define BLK      256
#define ROW_ITERS (FEAT / (BLK * 4))   // 392 float4 iterations per row
#define COL_BLKS  (FEAT / (BLK * 4))   // 392 column-chunk blocks

#define NEG_HUGE (-3.402823466e+38f)

typedef __attribute__((ext_vector_type(2))) float v2f;
typedef __attribute__((ext_vector_type(8))) float v8f;

// ---- online softmax state ops: m (running max), s0 = sum e^(x-m),
// ----                           s1 = sum (x-m) e^(x-m)
__device__ __forceinline__ void os_push(float& m, float& s0, float& s1, float x) {
    if (x > m) {                       // rare after warmup
        float d = fmaxf(m - x, -100.0f);   // clamp: avoids -inf*0 at init
        float a = __expf(d);
        float as0 = a * s0;
        s1 = a * s1 + d * as0;
        s0 = as0 + 1.0f;
        m  = x;
    } else {
        float d = x - m;
        float e = __expf(d);
        s0 += e;
        s1 = fmaf(d, e, s1);
    }
}

__device__ __forceinline__ void os_merge(float& m, float& s0, float& s1,
                                         float mb, float s0b, float s1b) {
    float mm = fmaxf(m, mb);
    float da = fmaxf(m  - mm, -100.0f);
    float db = fmaxf(mb - mm, -100.0f);
    float ea = __expf(da);
    float eb = __expf(db);
    s1 = ea * (s1 + da * s0) + eb * (s1b + db * s0b);
    s0 = ea * s0 + eb * s0b;
    m  = mm;
}

__device__ __forceinline__ void block_os_reduce(int tid, float& m, float& s0, float& s1,
                                                float* sm, float* ss0, float* ss1) {
    sm[tid] = m; ss0[tid] = s0; ss1[tid] = s1;
    __syncthreads();
    for (int off = BLK / 2; off > 0; off >>= 1) {
        if (tid < off) {
            os_merge(m, s0, s1, sm[tid + off], ss0[tid + off], ss1[tid + off]);
            sm[tid] = m; ss0[tid] = s0; ss1[tid] = s1;
        }
        __syncthreads();
    }
}

// ---------------- pass 1: per-row online softmax entropy -------------------
__global__ void row_stats_kernel(const float* __restrict__ x,
                                 float* __restrict__ row_ent) {
    const int tid = threadIdx.x;
    const int row = blockIdx.x;
    const float4* rp = (const float4*)(x + (size_t)row * FEAT);

    float m = NEG_HUGE, s0 = 0.0f, s1 = 0.0f;
    for (int it = 0; it < ROW_ITERS; ++it) {
        if (it + 16 < ROW_ITERS)   // gfx1250 global_prefetch_b8
            __builtin_prefetch(&rp[tid + (it + 16) * BLK], 0, 1);
        float4 v = rp[tid + it * BLK];
        os_push(m, s0, s1, v.x);
        os_push(m, s0, s1, v.y);
        os_push(m, s0, s1, v.z);
        os_push(m, s0, s1, v.w);
    }

    __shared__ float sm[BLK], ss0[BLK], ss1[BLK];
    block_os_reduce(tid, m, s0, s1, sm, ss0, ss1);
    if (tid == 0)
        row_ent[row] = logf(s0) - s1 / s0;   // entropy of this row
}

// ------- pass 2: column means (coalesced) + softmax partials of mean -------
__global__ void col_stats_kernel(const float* __restrict__ x,
                                 float* __restrict__ part /* [3*COL_BLKS] */) {
    const int tid = threadIdx.x;
    const size_t j = ((size_t)blockIdx.x * BLK + tid) * 4;

    float4 acc = make_float4(0.f, 0.f, 0.f, 0.f);
    for (int t = 0; t < T_ROWS; ++t) {
        if (t + 4 < T_ROWS)
            __builtin_prefetch(x + (size_t)(t + 4) * FEAT + j, 0, 1);
        float4 v = *(const float4*)(x + (size_t)t * FEAT + j);
        acc.x += v.x; acc.y += v.y; acc.z += v.z; acc.w += v.w;
    }
    const float inv = 1.0f / (float)T_ROWS;
    float a0 = acc.x * inv, a1 = acc.y * inv, a2 = acc.z * inv, a3 = acc.w * inv;

    // exact local softmax stats over this thread's 4 mean values
    float m = fmaxf(fmaxf(a0, a1), fmaxf(a2, a3));
    float e0 = __expf(a0 - m), e1 = __expf(a1 - m);
    float e2 = __expf(a2 - m), e3 = __expf(a3 - m);
    float s0 = e0 + e1 + e2 + e3;
    float s1 = (a0 - m) * e0 + (a1 - m) * e1 + (a2 - m) * e2 + (a3 - m) * e3;

    __shared__ float sm[BLK], ss0[BLK], ss1[BLK];
    block_os_reduce(tid, m, s0, s1, sm, ss0, ss1);
    if (tid == 0) {
        part[blockIdx.x]                = m;
        part[COL_BLKS + blockIdx.x]     = s0;
        part[2 * COL_BLKS + blockIdx.x] = s1;
    }
}

// ----------------- pass 3: finalize (single block, uses WMMA) --------------
__global__ void finalize_kernel(const float* __restrict__ row_ent,
                                const float* __restrict__ part,
                                float* __restrict__ out) {
    const int tid = threadIdx.x;

    // --- L2: merge the 392 column-mean partials ---
    float m = part[tid];
    float s0 = part[COL_BLKS + tid];
    float s1 = part[2 * COL_BLKS + tid];
    int j2 = tid + BLK;
    if (j2 < COL_BLKS)
        os_merge(m, s0, s1, part[j2], part[COL_BLKS + j2], part[2 * COL_BLKS + j2]);

    __shared__ float sm[BLK], ss0[BLK], ss1[BLK];
    block_os_reduce(tid, m, s0, s1, sm, ss0, ss1);
    if (tid == 0)
        out[1] = s1 / s0 - logf(s0);     // L2 = -entropy(mean vector)

    // --- L1: mean of 256 row entropies via V_WMMA_F32_16X16X4_F32 ---
    // A = (1/256) * ones(16x4)  =>  D[m][n] = (1/256) * sum_k B[k][n] + C.
    // A uniform => result invariant to B's lane->(k,n) mapping; 4 tiles cover
    // all 256 entropies. D's rows are identical (M=0 in lanes 0-15, M=8 in
    // lanes 16-31), so full-wave sum of D vgpr0 = 2 * L1.
    if (tid < 32) {                      // exactly wave 0: EXEC all ones
        v2f a; a[0] = 1.0f / 256.0f; a[1] = 1.0f / 256.0f;
        v8f c = {};
        for (int i = 0; i < 4; ++i) {
            v2f b;
            b[0] = row_ent[i * 64 + tid];
            b[1] = row_ent[i * 64 + 32 + tid];
            c = __builtin_amdgcn_wmma_f32_16x16x4_f32(
                    /*neg_a=*/false, a, /*neg_b=*/false, b,
                    /*c_mod=*/(short)0, c, /*reuse_a=*/false, /*reuse_b=*/false);
        }
        float v = c[0];
        for (int off = 16; off > 0; off >>= 1)
            v += __shfl_xor(v, off, 32);
        if (tid == 0)
            out[0] = 0.5f * v;           // L1
    }
}

extern "C" void kernel_launch(void* const* d_in, const int* in_sizes, int n_in,
                              void* d_out, int out_size, void* d_ws, size_t ws_size,
                              hipStream_t stream) {
    const float* x = (const float*)d_in[0];
    float* out = (float*)d_out;
    float* ws = (float*)d_ws;

    float* row_ent = ws;            // 256 floats
    float* part    = ws + 256;      // 3 * 392 floats

    row_stats_kernel<<<T_ROWS, BLK, 0, stream>>>(x, row_ent);
    col_stats_kernel<<<COL_BLKS, BLK, 0, stream>>>(x, part);
    finalize_kernel<<<1, BLK, 0, stream>>>(row_ent, part, out);
}